// ResGATEGraph_89172111000294
// MI455X (gfx1250) — compile-verified
//
#include <hip/hip_runtime.h>
#include <hip/hip_bf16.h>

// ---------------------------------------------------------------------------
// RGCN (basis decomp, per-(rel,dst) mean) + ResGatedGraphConv for MI455X.
//
// v2 changes vs v1 (driven by disasm): weight matrices are pre-swizzled into
// WMMA-fragment order so each B fragment is ONE coalesced global_load_b64 per
// lane (v1 needed two strided b32 loads), and each wave now carries TWO 16-row
// A tiles sharing every B fragment -> ~10 VMEM issues per 16 WMMAs instead of
// 17 per 8.  All matrix math stays on v_wmma_f32_16x16x4_f32 (fp32 fidelity;
// roofline: ~28 GFLOP fp32-WMMA ~ the ~50us of L2/HBM edge traffic).
// Workspace: ~107.5 MB of d_ws.
// ---------------------------------------------------------------------------

typedef __attribute__((ext_vector_type(2))) float v2f;
typedef __attribute__((ext_vector_type(8))) float v8f;

#define GD 128   // == H1 == H2 in this model

__device__ __forceinline__ v8f wmma_f32(v2f a, v2f b, v8f c) {
  return __builtin_amdgcn_wmma_f32_16x16x4_f32(false, a, false, b, (short)0, c,
                                               false, false);
}

// Fragment swizzle position for element (i=K-row, o=N-col) of a 128x128 B:
//   fragment (ks = i/4, nt = o/16); within fragment lane = (i&3)/2*16 + (o&15),
//   component = (i&3)&1.  One fragment = 32 lanes * float2 = 256B contiguous.
__device__ __forceinline__ int swz_pos(int i, int o) {
  int ks = i >> 2, rr = i & 3;
  int lane = ((rr >> 1) << 4) + (o & 15);
  int nt = o >> 4;
  return (((ks * 8 + nt) * 32) + lane) * 2 + (rr & 1);
}

// ---------------- small utility kernels ----------------

__global__ void k_zero_i32(int* __restrict__ p, int count) {
  int i = blockIdx.x * blockDim.x + threadIdx.x;
  if (i < count) p[i] = 0;
}

// row-major 128x128 -> WMMA fragment order (for the 5 dense weight inputs)
__global__ void k_swizzle(const float* __restrict__ Bm, float* __restrict__ Bs) {
  int idx = blockIdx.x * blockDim.x + threadIdx.x;
  if (idx >= GD * GD) return;
  int i = idx >> 7, o = idx & 127;
  Bs[swz_pos(i, o)] = Bm[idx];
}

// W[r,i,o] = sum_b comp[r,b]*basis[b,i,o], stored directly in swizzled order
__global__ void k_weights(const float* __restrict__ comp,
                          const float* __restrict__ basis,
                          float* __restrict__ Wsw, int r_count, int b_count) {
  int idx = blockIdx.x * blockDim.x + threadIdx.x;
  int total = r_count * GD * GD;
  if (idx >= total) return;
  int r  = idx / (GD * GD);
  int io = idx % (GD * GD);
  float acc = 0.f;
  for (int b = 0; b < b_count; ++b)
    acc += comp[r * b_count + b] * basis[b * GD * GD + io];
  Wsw[(size_t)r * GD * GD + swz_pos(io >> 7, io & 127)] = acc;
}

// per-(relation,dst) counts + per-relation bucket sizes
__global__ void k_hist(const int* __restrict__ etype, const int* __restrict__ dstA,
                       int* __restrict__ cnt, int* __restrict__ bcnt,
                       int n, int e_count) {
  int e = blockIdx.x * blockDim.x + threadIdx.x;
  if (e >= e_count) return;
  int t = etype[e];
  atomicAdd(&cnt[t * n + dstA[e]], 1);
  atomicAdd(&bcnt[t], 1);
}

// 8-entry scans: bucket offsets, scatter cursors, cumulative 32-edge tile counts
__global__ void k_scan(const int* __restrict__ bcnt, int* __restrict__ boff,
                       int* __restrict__ cursor, int* __restrict__ tcum,
                       int r_count) {
  if (blockIdx.x == 0 && threadIdx.x == 0) {
    int o = 0, tc = 0;
    boff[0] = 0; tcum[0] = 0;
    for (int r = 0; r < r_count; ++r) {
      cursor[r] = o;
      o += bcnt[r];              boff[r + 1] = o;
      tc += (bcnt[r] + 31) >> 5; tcum[r + 1] = tc;
    }
  }
}

// counting-sort scatter: edge ids grouped by relation
__global__ void k_scatter(const int* __restrict__ etype, int* __restrict__ cursor,
                          int* __restrict__ sorted, int e_count) {
  int e = blockIdx.x * blockDim.x + threadIdx.x;
  if (e >= e_count) return;
  int p = atomicAdd(&cursor[etype[e]], 1);
  sorted[p] = e;
}

// ---------------- wave GEMM: Y[32x128] = X[32x128] @ B[128x128] + bias
// One wave per 32-row tile; two A fragments share every swizzled B fragment.
__global__ __launch_bounds__(32) void k_rowgemm_bias(
    const float* __restrict__ X, const float* __restrict__ Bswz,
    const float* __restrict__ bias, float* __restrict__ Y, int nrows) {
  int row0 = blockIdx.x * 32;
  if (row0 >= nrows) return;
  int lane = threadIdx.x;
  int r0 = row0 + (lane & 15);
  int r1 = row0 + 16 + (lane & 15);
  if (r0 >= nrows) r0 = nrows - 1;              // tail-safe (dups not stored)
  if (r1 >= nrows) r1 = nrows - 1;
  const float* x0 = X + (size_t)r0 * GD;
  const float* x1 = X + (size_t)r1 * GD;
  const v2f* Bf = (const v2f*)Bswz;

  v8f acc0[8] = {}, acc1[8] = {};
#pragma unroll
  for (int ks = 0; ks < 32; ++ks) {             // ks = k/4
    int ka = ks * 4 + 2 * (lane >> 4);
    v2f a0; a0.x = x0[ka]; a0.y = x0[ka + 1];
    v2f a1; a1.x = x1[ka]; a1.y = x1[ka + 1];
#pragma unroll
    for (int nt = 0; nt < 8; ++nt) {
      v2f bfr = Bf[(ks * 8 + nt) * 32 + lane];  // one coalesced b64 per lane
      acc0[nt] = wmma_f32(a0, bfr, acc0[nt]);
      acc1[nt] = wmma_f32(a1, bfr, acc1[nt]);
    }
  }
  int nlo = lane & 15, mhi = (lane >> 4) * 8;
#pragma unroll
  for (int nt = 0; nt < 8; ++nt) {
    int ncol = nt * 16 + nlo;
    float bv = bias[ncol];
#pragma unroll
    for (int j = 0; j < 8; ++j) {
      int m0 = row0 + mhi + j;
      int m1 = row0 + 16 + mhi + j;
      if (m0 < nrows) Y[(size_t)m0 * GD + ncol] = acc0[nt][j] + bv;
      if (m1 < nrows) Y[(size_t)m1 * GD + ncol] = acc1[nt][j] + bv;
    }
  }
}

// ---------------- RGCN messages: 32 same-relation edges per wave.
// h[dst] += (1/cnt[r,dst]) * (x[src] @ W_r)   via WMMA + scaled f32 atomics.
__global__ __launch_bounds__(32) void k_msg(
    const float* __restrict__ X, const float* __restrict__ Wsw,
    const int* __restrict__ srcA, const int* __restrict__ dstA,
    const int* __restrict__ sorted, const int* __restrict__ boff,
    const int* __restrict__ tcum, const int* __restrict__ cnt,
    float* __restrict__ h, int n, int r_count) {
  __shared__ int   s_srcoff[32];
  __shared__ int   s_dst[32];
  __shared__ float s_scale[32];

  int t = blockIdx.x;
  if (t >= tcum[r_count]) return;               // wave-uniform exit
  int r = 0;
  for (int i = 1; i <= 8; ++i) if (i <= r_count && t >= tcum[i]) r = i;

  int base = boff[r] + (t - tcum[r]) * 32;
  int nE   = boff[r + 1] - base;  if (nE > 32) nE = 32;

  int lane = threadIdx.x;
  {
    int row = lane;                             // rows 0..31 of the edge tile
    int ei  = base + (row < nE ? row : nE - 1); // clamp: in-bounds pad loads
    int eid = sorted[ei];
    int s   = srcA[eid];
    int d   = dstA[eid];
    int c   = cnt[r * n + d];
    s_srcoff[row] = s * GD;
    s_dst[row]    = d;
    s_scale[row]  = (row < nE) ? (1.0f / (float)(c < 1 ? 1 : c)) : 0.0f;
  }
  __syncthreads();

  const v2f*  Bf = (const v2f*)(Wsw + (size_t)r * GD * GD);
  const float* x0 = X + s_srcoff[lane & 15];
  const float* x1 = X + s_srcoff[16 + (lane & 15)];

  v8f acc0[8] = {}, acc1[8] = {};
#pragma unroll
  for (int ks = 0; ks < 32; ++ks) {
    int ka = ks * 4 + 2 * (lane >> 4);
    v2f a0; a0.x = x0[ka]; a0.y = x0[ka + 1];
    v2f a1; a1.x = x1[ka]; a1.y = x1[ka + 1];
#pragma unroll
    for (int nt = 0; nt < 8; ++nt) {
      v2f bfr = Bf[(ks * 8 + nt) * 32 + lane];
      acc0[nt] = wmma_f32(a0, bfr, acc0[nt]);
      acc1[nt] = wmma_f32(a1, bfr, acc1[nt]);
    }
  }
  int nlo = lane & 15, mhi = (lane >> 4) * 8;
#pragma unroll
  for (int nt = 0; nt < 8; ++nt) {
    int ncol = nt * 16 + nlo;
#pragma unroll
    for (int j = 0; j < 8; ++j) {
      int m = mhi + j;
      float sc0 = s_scale[m];
      float sc1 = s_scale[16 + m];
      if (sc0 != 0.0f)
        atomicAdd(&h[(size_t)s_dst[m] * GD + ncol], acc0[nt][j] * sc0);
      if (sc1 != 0.0f)
        atomicAdd(&h[(size_t)s_dst[16 + m] * GD + ncol], acc1[nt][j] * sc1);
    }
  }
}

// ---------------- gate phase: out[dst] += sigmoid(k[dst]+q[src]) * v[src]
// One wave per edge, float4 per lane; k/q/v/out all L2-resident (102MB<192MB).
__global__ __launch_bounds__(256) void k_gate(
    const int* __restrict__ srcA, const int* __restrict__ dstA,
    const float* __restrict__ kb, const float* __restrict__ qb,
    const float* __restrict__ vb, float* __restrict__ out, int e_count) {
  int e = blockIdx.x * 8 + (threadIdx.x >> 5);
  if (e >= e_count) return;
  int lane = threadIdx.x & 31;
  size_t s = (size_t)srcA[e] * GD, d = (size_t)dstA[e] * GD;
  int c = lane * 4;
  float4 kv = *(const float4*)(kb + d + c);
  float4 qv = *(const float4*)(qb + s + c);
  float4 vv = *(const float4*)(vb + s + c);
  float4 g;
  g.x = vv.x / (1.0f + __expf(-(kv.x + qv.x)));
  g.y = vv.y / (1.0f + __expf(-(kv.y + qv.y)));
  g.z = vv.z / (1.0f + __expf(-(kv.z + qv.z)));
  g.w = vv.w / (1.0f + __expf(-(kv.w + qv.w)));
  atomicAdd(&out[d + c + 0], g.x);
  atomicAdd(&out[d + c + 1], g.y);
  atomicAdd(&out[d + c + 2], g.z);
  atomicAdd(&out[d + c + 3], g.w);
}

// ---------------- host orchestration ----------------

extern "C" void kernel_launch(void* const* d_in, const int* in_sizes, int n_in,
                              void* d_out, int out_size, void* d_ws, size_t ws_size,
                              hipStream_t stream) {
  const float* x      = (const float*)d_in[0];
  const int*   eidx   = (const int*)d_in[1];
  // d_in[2] = edge_norm: unused by the reference
  const int*   etype  = (const int*)d_in[3];
  const float* basis  = (const float*)d_in[4];
  const float* comp   = (const float*)d_in[5];
  const float* root   = (const float*)d_in[6];
  const float* bias1  = (const float*)d_in[7];
  const float* w_key  = (const float*)d_in[8];
  const float* b_key  = (const float*)d_in[9];
  const float* w_qry  = (const float*)d_in[10];
  const float* b_qry  = (const float*)d_in[11];
  const float* w_val  = (const float*)d_in[12];
  const float* b_val  = (const float*)d_in[13];
  const float* w_skip = (const float*)d_in[14];
  const float* b_skip = (const float*)d_in[15];
  float* out = (float*)d_out;

  const int n = in_sizes[0] / GD;             // 50000
  const int e = in_sizes[3];                  // 600000
  const int b = in_sizes[4] / (GD * GD);      // 32
  const int r = in_sizes[5] / b;              // 8

  const int* srcA = eidx;
  const int* dstA = eidx + e;

  // workspace carving (~107.5 MB)
  float* Wsw   = (float*)d_ws;                      // r*128*128 (swizzled)
  float* rootS = Wsw   + (size_t)r * GD * GD;       // 5 swizzled dense weights
  float* keyS  = rootS + GD * GD;
  float* qryS  = keyS  + GD * GD;
  float* valS  = qryS  + GD * GD;
  float* skipS = valS  + GD * GD;
  float* h     = skipS + GD * GD;                   // n*128
  float* kbuf  = h     + (size_t)n * GD;
  float* qbuf  = kbuf  + (size_t)n * GD;
  float* vbuf  = qbuf  + (size_t)n * GD;
  int* cnt     = (int*)(vbuf + (size_t)n * GD);     // r*n
  int* bcnt    = cnt   + (size_t)r * n;             // r
  int* boff    = bcnt  + r;                         // r+1
  int* cursor  = boff  + r + 1;                     // r
  int* tcum    = cursor + r;                        // r+1
  int* sorted  = tcum  + r + 1;                     // e

  // 1. zero counters (cnt + bcnt contiguous)
  int zc = r * n + r;
  k_zero_i32<<<(zc + 255) / 256, 256, 0, stream>>>(cnt, zc);
  // 2. fold basis decomposition into swizzled relation weights
  k_weights<<<(r * GD * GD + 255) / 256, 256, 0, stream>>>(comp, basis, Wsw, r, b);
  // 2b. swizzle the 5 dense weights into fragment order
  int sg = (GD * GD + 255) / 256;
  k_swizzle<<<sg, 256, 0, stream>>>(root,   rootS);
  k_swizzle<<<sg, 256, 0, stream>>>(w_key,  keyS);
  k_swizzle<<<sg, 256, 0, stream>>>(w_qry,  qryS);
  k_swizzle<<<sg, 256, 0, stream>>>(w_val,  valS);
  k_swizzle<<<sg, 256, 0, stream>>>(w_skip, skipS);
  // 3-5. counting-sort edges by relation; per-(r,dst) counts
  k_hist<<<(e + 255) / 256, 256, 0, stream>>>(etype, dstA, cnt, bcnt, n, e);
  k_scan<<<1, 1, 0, stream>>>(bcnt, boff, cursor, tcum, r);
  k_scatter<<<(e + 255) / 256, 256, 0, stream>>>(etype, cursor, sorted, e);
  // 6. h = x @ root + bias1   (writes all of h, so no zero-init needed)
  int mtiles = (n + 31) / 32;
  k_rowgemm_bias<<<mtiles, 32, 0, stream>>>(x, rootS, bias1, h, n);
  // 7. h += per-(r,dst)-mean messages (32 same-relation edges per wave)
  int maxT = (e + 31) / 32 + r;
  k_msg<<<maxT, 32, 0, stream>>>(x, Wsw, srcA, dstA, sorted, boff, tcum, cnt, h, n, r);
  // 8. k/q/v into ws, skip term written (not added) straight into d_out
  k_rowgemm_bias<<<mtiles, 32, 0, stream>>>(h, keyS,  b_key,  kbuf, n);
  k_rowgemm_bias<<<mtiles, 32, 0, stream>>>(h, qryS,  b_qry,  qbuf, n);
  k_rowgemm_bias<<<mtiles, 32, 0, stream>>>(h, valS,  b_val,  vbuf, n);
  k_rowgemm_bias<<<mtiles, 32, 0, stream>>>(h, skipS, b_skip, out,  n);
  // 9. gated scatter-add (L2-resident gathers + f32 atomics)
  k_gate<<<(e + 7) / 8, 256, 0, stream>>>(srcA, dstA, kbuf, qbuf, vbuf, out, e);
}